// Conv2d_88682484728535
// MI455X (gfx1250) — compile-verified
//
#include <hip/hip_runtime.h>
#include <stdint.h>
#include <stddef.h>

// FHE BSGS rotate/plaintext-mul/accumulate, bandwidth-bound fp32.
// out[b,s] = sum_t x[b,(s+roll+2^t*stride) mod S] * diag[t,(s+roll) mod S]
// with roll = S/(2*reps).  S and reps are powers of two (FHE slot math), so
// every "mod S" is AND(S-1) and the roll division is a shift.

#define NUM_T   16
#define TILE_S  1024      // output slots per block
#define THREADS 256       // TILE_S / 4
#define BPB     4         // batches per block

typedef float f4 __attribute__((ext_vector_type(4)));
typedef int   vi4 __attribute__((vector_size(16)));   // matches builtin's V4i

__device__ __forceinline__ void async_copy_f4_to_lds(const float* gsrc, float* ldst) {
#if defined(__HIP_DEVICE_COMPILE__) && __has_builtin(__builtin_amdgcn_global_load_async_to_lds_b128)
  // per-lane 16B async copy: global -> LDS, tracked by ASYNCcnt
  __builtin_amdgcn_global_load_async_to_lds_b128(
      (__attribute__((address_space(1))) vi4*)gsrc,
      (__attribute__((address_space(3))) vi4*)ldst,
      0, 0);
#else
  *(f4*)ldst = *(const f4*)gsrc;
#endif
}

__device__ __forceinline__ void wait_async_zero() {
#if defined(__HIP_DEVICE_COMPILE__)
#if __has_builtin(__builtin_amdgcn_s_wait_asynccnt)
  __builtin_amdgcn_s_wait_asynccnt(0);
#else
  asm volatile("s_wait_asynccnt 0" ::: "memory");
#endif
#endif
}

__global__ __launch_bounds__(THREADS) void fhe_bsgs_rotmul_acc(
    const float* __restrict__ x,      // [B, S]
    const float* __restrict__ diag,   // [NUM_T, S]
    const int*   __restrict__ pstride,
    const int*   __restrict__ preps,
    float*       __restrict__ out,    // [B, S]
    int S, int B)
{
  __shared__ __align__(16) float dtile[NUM_T * TILE_S];   // 64 KB

  const unsigned tid  = threadIdx.x;
  const unsigned uS   = (unsigned)S;
  const unsigned mask = uS - 1u;                 // S is a power of two
  const int stride = *pstride;
  const unsigned reps = (unsigned)(*preps);      // power of two
  // roll = S / (2*reps) without integer division
  const unsigned roll = ((uS >> 1) >> __builtin_ctz(reps)) & mask;

  const unsigned s_base  = (unsigned)blockIdx.x * TILE_S;
  const unsigned sp_base = (s_base + roll) & mask;        // pre-roll slot of tile start
  const int b0 = (int)blockIdx.y * BPB;
  const unsigned col = tid * 4u;                          // thread's 4 slots in tile

  // rotation amounts 2^t * stride (mod S): for pow2 S this is shift+mask
  unsigned shifts[NUM_T];
#pragma unroll
  for (int t = 0; t < NUM_T; ++t)
    shifts[t] = (((unsigned)stride) << t) & mask;

  // ---- stage diagonal tile into LDS (async fast path) ----
  const unsigned g0 = (sp_base + col) & mask;             // thread's gather base for diag
  if ((g0 & 3u) == 0u) {                                  // aligned, wrap-free 16B chunks
#pragma unroll
    for (int t = 0; t < NUM_T; ++t)
      async_copy_f4_to_lds(diag + (size_t)t * uS + g0, &dtile[t * TILE_S + col]);
    wait_async_zero();
  } else {                                                // generic wrap-safe path
#pragma unroll
    for (int t = 0; t < NUM_T; ++t)
      for (int i = 0; i < 4; ++i)
        dtile[t * TILE_S + col + i] = diag[(size_t)t * uS + ((sp_base + col + i) & mask)];
  }
  __syncthreads();

  // gather bases are batch-independent: precompute once
  unsigned gidx[NUM_T];
#pragma unroll
  for (int t = 0; t < NUM_T; ++t)
    gidx[t] = (g0 + shifts[t]) & mask;

#pragma unroll 1   // keep the batch loop rolled: compact code, same dynamic work
  for (int bb = 0; bb < BPB; ++bb) {
    const int b = b0 + bb;
    if (b >= B) break;
    const float* __restrict__ xb = x + (size_t)b * uS;

    float ax = 0.f, ay = 0.f, az = 0.f, aw = 0.f;
#pragma unroll
    for (int t = 0; t < NUM_T; ++t) {
      const unsigned g = gidx[t];
      f4 xv;
      if ((g & 3u) == 0u) {                 // wave-uniform branch (g%4 uniform)
        xv = *(const f4*)(xb + g);          // global_load_b128
      } else {                              // shifts 1,2: scalar wrap-safe gather
        xv.x = xb[g];
        xv.y = xb[(g + 1u) & mask];
        xv.z = xb[(g + 2u) & mask];
        xv.w = xb[(g + 3u) & mask];
      }
      const f4 dv = *(const f4*)&dtile[t * TILE_S + col];  // ds_load_b128
      ax = fmaf(xv.x, dv.x, ax);
      ay = fmaf(xv.y, dv.y, ay);
      az = fmaf(xv.z, dv.z, az);
      aw = fmaf(xv.w, dv.w, aw);
    }
    f4 res; res.x = ax; res.y = ay; res.z = az; res.w = aw;
    // output never re-read by us: non-temporal wide store
    __builtin_nontemporal_store(res, (f4*)(out + (size_t)b * uS + s_base + col));
  }
}

extern "C" void kernel_launch(void* const* d_in, const int* in_sizes, int n_in,
                              void* d_out, int out_size, void* d_ws, size_t ws_size,
                              hipStream_t stream) {
  const float* x       = (const float*)d_in[0];
  const float* diag    = (const float*)d_in[1];
  const int*   pstride = (const int*)d_in[2];
  const int*   preps   = (const int*)d_in[3];
  float*       out     = (float*)d_out;

  const int T = NUM_T;
  const int S = in_sizes[1] / T;     // 65536
  const int B = in_sizes[0] / S;     // 64

  dim3 grid((unsigned)(S / TILE_S), (unsigned)((B + BPB - 1) / BPB));
  fhe_bsgs_rotmul_acc<<<grid, THREADS, 0, stream>>>(x, diag, pstride, preps, out, S, B);
}